// SMFNet_38577396252906
// MI455X (gfx1250) — compile-verified
//
#include <hip/hip_runtime.h>

typedef float v2f __attribute__((ext_vector_type(2)));
typedef float v4f __attribute__((ext_vector_type(4)));
typedef float v8f __attribute__((ext_vector_type(8)));

#define TILE_ROWS 16
#define WAVES_PER_BLOCK 8
#define ROWS_PER_BLOCK (TILE_ROWS * WAVES_PER_BLOCK)
#define D 128

// One wave per 16-row tile.
//   D0 = X[base:base+16] @ theta[base:base+16]^T      (diag -> w_diag)
//   D1 = X[base:base+16] @ theta[base+1:base+17]^T    (diag -> w_sup, halo row included)
// V = relu(X @ xi^T + g_bias) accumulated inline off the same A-fragment loads.
__global__ __launch_bounds__(256) void smf_v0_kernel(
    const float* __restrict__ X,
    const float* __restrict__ theta,
    const float* __restrict__ f_bias,
    const float* __restrict__ xi,
    const float* __restrict__ g_bias,
    float* __restrict__ out,
    int n)
{
    __shared__ __align__(16) float s_xi[2 * D];   // xi[0][:], xi[1][:]
    if (threadIdx.x < 2 * D) s_xi[threadIdx.x] = xi[threadIdx.x];
    __syncthreads();

    const int lane = threadIdx.x & 31;
    const int wave = threadIdx.x >> 5;
    const long nm1 = (long)n - 1;
    const long base = ((long)blockIdx.x * WAVES_PER_BLOCK + wave) * TILE_ROWS;

    const int m = lane & 15;   // A: row-in-tile (M); B: col-in-tile (N)
    const int h = lane >> 4;   // lane half selects which K pair this lane feeds

    const long rowA  = base + m;                 // X row / theta row for D0
    const long rowB1 = (base + 1 + m) & nm1;     // theta row for D1 (wraps at N)

    const v4f* __restrict__ Xp   = (const v4f*)(X     + rowA  * D + 4 * h);
    const v4f* __restrict__ T0p  = (const v4f*)(theta + rowA  * D + 4 * h);
    const v4f* __restrict__ T1p  = (const v4f*)(theta + rowB1 * D + 4 * h);
    const v4f* __restrict__ xi0p = (const v4f*)(s_xi + 4 * h);
    const v4f* __restrict__ xi1p = (const v4f*)(s_xi + D + 4 * h);

    v8f c0 = {};    // D0 accumulator (16x16 f32)
    v8f c1 = {};    // D1 accumulator
    float v0 = 0.f, v1 = 0.f;   // partial X[m].xi dots over this lane's columns

    // K permutation: WMMA step (t, s) maps its K={2h+j} to column 8t + 4h + 2s + j.
    // Each b128 load feeds two WMMA k-steps; union over (t, s, h) covers all 128 cols.
    #pragma unroll 4
    for (int t = 0; t < 16; ++t) {
        v4f a  = Xp[2 * t];
        v4f b0 = T0p[2 * t];
        v4f b1 = T1p[2 * t];
        v4f x0 = xi0p[2 * t];
        v4f x1 = xi1p[2 * t];

        v2f aLo  = { a.x,  a.y  }, aHi  = { a.z,  a.w  };
        v2f b0Lo = { b0.x, b0.y }, b0Hi = { b0.z, b0.w };
        v2f b1Lo = { b1.x, b1.y }, b1Hi = { b1.z, b1.w };

        c0 = __builtin_amdgcn_wmma_f32_16x16x4_f32(false, aLo, false, b0Lo, (short)0, c0, false, false);
        c1 = __builtin_amdgcn_wmma_f32_16x16x4_f32(false, aLo, false, b1Lo, (short)0, c1, false, false);
        c0 = __builtin_amdgcn_wmma_f32_16x16x4_f32(false, aHi, false, b0Hi, (short)0, c0, false, false);
        c1 = __builtin_amdgcn_wmma_f32_16x16x4_f32(false, aHi, false, b1Hi, (short)0, c1, false, false);

        v0 = fmaf(a.x, x0.x, v0); v0 = fmaf(a.y, x0.y, v0);
        v0 = fmaf(a.z, x0.z, v0); v0 = fmaf(a.w, x0.w, v0);
        v1 = fmaf(a.x, x1.x, v1); v1 = fmaf(a.y, x1.y, v1);
        v1 = fmaf(a.z, x1.z, v1); v1 = fmaf(a.w, x1.w, v1);
    }

    // Halo V for row base+16 (needed by m==15's V[i+1]); full-wave cooperative dot.
    const long rowH = (base + TILE_ROWS) & nm1;
    float h0 = 0.f, h1 = 0.f;
    {
        v4f xh = *(const v4f*)(X + rowH * D + 4 * lane);
        v4f x0 = *(const v4f*)(s_xi + 4 * lane);
        v4f x1 = *(const v4f*)(s_xi + D + 4 * lane);
        h0 = fmaf(xh.x, x0.x, h0); h0 = fmaf(xh.y, x0.y, h0);
        h0 = fmaf(xh.z, x0.z, h0); h0 = fmaf(xh.w, x0.w, h0);
        h1 = fmaf(xh.x, x1.x, h1); h1 = fmaf(xh.y, x1.y, h1);
        h1 = fmaf(xh.z, x1.z, h1); h1 = fmaf(xh.w, x1.w, h1);
        #pragma unroll
        for (int off = 16; off > 0; off >>= 1) {
            h0 += __shfl_xor(h0, off, 32);
            h1 += __shfl_xor(h1, off, 32);
        }
    }

    // Complete per-row xi dots: lane L and L+16 hold complementary column halves.
    v0 += __shfl_xor(v0, 16, 32);
    v1 += __shfl_xor(v1, 16, 32);

    const float gb0 = g_bias[0];
    const float gb1 = g_bias[1];
    const float Vr0 = fmaxf(v0 + gb0, 0.f);   // V[base+m][0] (held at lanes m and m+16)
    const float Vr1 = fmaxf(v1 + gb1, 0.f);
    const float VH0 = fmaxf(h0 + gb0, 0.f);   // V[base+16][*] in every lane
    const float VH1 = fmaxf(h1 + gb1, 0.f);

    // V[base+m+1]: neighbor lane within each half; m==15 takes the halo value.
    float nv0 = __shfl(Vr0, lane + 1, 32);
    float nv1 = __shfl(Vr1, lane + 1, 32);
    if (m == 15) { nv0 = VH0; nv1 = VH1; }

    // Diagonal extraction from the 16x16 f32 C layout:
    //   VGPR r, lanes 0-15:  M=r,   N=lane      -> diag at lane r      (rows 0..7)
    //   VGPR r, lanes 16-31: M=8+r, N=lane-16   -> diag at lane 24+r   (rows 8..15)
    const bool active = (lane < 8) || (lane >= 24);
    const int  idx    = (lane < 8) ? lane : (lane - 24);
    float d0 = 0.f, d1 = 0.f;
    #pragma unroll
    for (int r = 0; r < 8; ++r) {
        if (idx == r) { d0 = c0[r]; d1 = c1[r]; }
    }

    if (active) {
        const long row  = base + ((lane < 8) ? lane : (lane - 16));
        const long rown = (row + 1) & nm1;
        const float wd = fmaxf(d0 + f_bias[row],  0.f);   // relu(X[i].theta[i]   + b[i])
        const float ws = fmaxf(d1 + f_bias[rown], 0.f);   // relu(X[i].theta[i+1] + b[i+1])
        out[row * 2 + 0] = wd * Vr0 + ws * nv0;
        out[row * 2 + 1] = wd * Vr1 + ws * nv1;
    }
}

extern "C" void kernel_launch(void* const* d_in, const int* in_sizes, int n_in,
                              void* d_out, int out_size, void* d_ws, size_t ws_size,
                              hipStream_t stream) {
    const float* X      = (const float*)d_in[0];
    const float* theta  = (const float*)d_in[1];
    const float* f_bias = (const float*)d_in[2];
    const float* xi     = (const float*)d_in[3];
    const float* g_bias = (const float*)d_in[4];
    float* out = (float*)d_out;

    const int n = in_sizes[2];              // N = 524288 (f_bias element count)
    const int blocks = n / ROWS_PER_BLOCK;  // 128 rows per 256-thread block (8 waves)
    smf_v0_kernel<<<blocks, 256, 0, stream>>>(X, theta, f_bias, xi, g_bias, out, n);
}